// TextDetectionLoss_38147899523462
// MI455X (gfx1250) — compile-verified
//
#include <hip/hip_runtime.h>
#include <math.h>

// ---------------------------------------------------------------------------
// MI455X text-detection loss.
//  - Bulk BCE+Dice reduction: 52.4 MB streamed -> ~2.3us at 23.3 TB/s (HBM bound)
//  - CDNA5 paths: global_load_async_to_lds_b128 double buffering (ASYNCcnt),
//    v_wmma_f32_16x16x4_f32 for the cross-wave block reduction.
// ---------------------------------------------------------------------------

typedef float v2f __attribute__((ext_vector_type(2)));
typedef float v8f __attribute__((ext_vector_type(8)));

#define TPB  256
#define NBLK 1280

// Async copy of 16 bytes/lane from global (SADDR + 32-bit byte offset, GVS mode)
// into this lane's LDS slot. Tracked by ASYNCcnt.
__device__ __forceinline__ void async_copy_b128(unsigned lds_off, unsigned byte_off,
                                                const void* base) {
  asm volatile("global_load_async_to_lds_b128 %0, %1, %2"
               :: "v"(lds_off), "v"(byte_off), "s"(base) : "memory");
}

__global__ __launch_bounds__(TPB)
void tmap_kernel(const float* __restrict__ p, const float* __restrict__ t,
                 float* __restrict__ ws, long long nvec) {
  __shared__ float4 shp[2][TPB];
  __shared__ float4 sht[2][TPB];
  __shared__ float  red[8][4];   // 8 waves x {bce, p*t, p, t}

  const int tid = threadIdx.x;
  long long i = (long long)blockIdx.x * TPB + tid;
  const long long stride = (long long)gridDim.x * TPB;

  float sb = 0.f, spt = 0.f, sp = 0.f, st = 0.f;

  bool have = i < nvec;
  int buf = 0;
  if (have) {
    async_copy_b128((unsigned)(size_t)&shp[0][tid], (unsigned)(i * 16), p);
    async_copy_b128((unsigned)(size_t)&sht[0][tid], (unsigned)(i * 16), t);
  }
  while (have) {
    long long nxt = i + stride;
    bool have_nxt = nxt < nvec;
    if (have_nxt) {
      async_copy_b128((unsigned)(size_t)&shp[buf ^ 1][tid], (unsigned)(nxt * 16), p);
      async_copy_b128((unsigned)(size_t)&sht[buf ^ 1][tid], (unsigned)(nxt * 16), t);
      asm volatile("s_wait_asynccnt 2" ::: "memory");   // oldest stage complete
    } else {
      asm volatile("s_wait_asynccnt 0" ::: "memory");
    }
    float4 pv = shp[buf][tid];
    float4 tv = sht[buf][tid];
    float pp[4] = {pv.x, pv.y, pv.z, pv.w};
    float tt[4] = {tv.x, tv.y, tv.z, tv.w};
#pragma unroll
    for (int k = 0; k < 4; ++k) {
      float pe = pp[k], te = tt[k];
      float lp  = fmaxf(__logf(pe),       -100.f);
      float l1p = fmaxf(__logf(1.f - pe), -100.f);
      sb  -= te * lp + (1.f - te) * l1p;
      spt += pe * te;
      sp  += pe;
      st  += te;
    }
    buf ^= 1; i = nxt; have = have_nxt;
  }

  // wave32 reduction of the four partials
#pragma unroll
  for (int off = 16; off > 0; off >>= 1) {
    sb  += __shfl_xor(sb,  off);
    spt += __shfl_xor(spt, off);
    sp  += __shfl_xor(sp,  off);
    st  += __shfl_xor(st,  off);
  }
  const int lane = tid & 31, wave = tid >> 5;
  if (lane == 0) { red[wave][0] = sb; red[wave][1] = spt; red[wave][2] = sp; red[wave][3] = st; }
  __syncthreads();

  // Cross-wave block reduction via WMMA: D = Ones(16x4) x B(4x16) + C, so
  // D[0][n] = sum_k B[k][n]. Put wave-k's partial for quantity n at B[k][n];
  // two chained wmma ops fold all 8 waves. Lanes 0..3 of D vgpr0 hold totals.
  if (tid < 32) {                       // whole wave 0: EXEC all ones for WMMA
    const int col = lane & 15, half = lane >> 4;
    const float m = (col < 4) ? 1.f : 0.f;
    v2f a;  a.x = 1.0f; a.y = 1.0f;     // A = ones
    v2f b0, b1;
    b0.x = red[half][col & 3]     * m;  // B row K=half   (lanes 0-15 K0, 16-31 K1)
    b0.y = red[2 + half][col & 3] * m;  // B row K=2+half
    b1.x = red[4 + half][col & 3] * m;
    b1.y = red[6 + half][col & 3] * m;
    v8f c = {};
    c = __builtin_amdgcn_wmma_f32_16x16x4_f32(false, a, false, b0, (short)0, c, false, false);
    c = __builtin_amdgcn_wmma_f32_16x16x4_f32(false, a, false, b1, (short)0, c, false, false);
    if (lane < 4) atomicAdd(&ws[lane], c[0]);
  }
}

// 1024 boxes total (B=16, N=64): gather-sample conf/bbox, IoU + sampled BCE,
// reduce, then combine with the bulk sums and emit the 4 outputs.
__global__ __launch_bounds__(1024)
void box_kernel(const float* __restrict__ conf, const float* __restrict__ bbox,
                const float* __restrict__ boxes, const float* __restrict__ ws,
                float* __restrict__ out, float ntotf, float inv_bn) {
  const int tid = threadIdx.x;          // 0..1023
  const int b = tid >> 6;
  const int n = tid & 63;
  const float* bx = boxes + ((size_t)b * 64 + n) * 5;
  float tc = bx[0];
  float t0 = bx[1], t1 = bx[2], t2 = bx[3], t3 = bx[4];
  int cy = (int)((t1 + t3) * 0.5f * 640.f);
  int cx = (int)((t0 + t2) * 0.5f * 640.f);
  cy = min(max(cy, 0), 639);
  cx = min(max(cx, 0), 639);
  const size_t hw  = (size_t)640 * 640;
  const size_t pix = (size_t)cy * 640 + cx;
  float cf = conf[(size_t)b * hw + pix];
  const float* bb = bbox + (size_t)b * 4 * hw + pix;
  float p0 = bb[0], p1 = bb[hw], p2 = bb[2 * hw], p3 = bb[3 * hw];

  float pa = (p2 - p0) * (p3 - p1);
  float ta = (t2 - t0) * (t3 - t1);
  float iw = fminf(p2, t2) - fmaxf(p0, t0);
  float ih = fminf(p3, t3) - fmaxf(p1, t1);
  float inter = fmaxf(iw, 0.f) * fmaxf(ih, 0.f);
  float iou = inter / (pa + ta - inter + 1e-6f);

  float lp  = fmaxf(__logf(cf),       -100.f);
  float l1p = fmaxf(__logf(1.f - cf), -100.f);
  float bce = -(tc * lp + (1.f - tc) * l1p);

#pragma unroll
  for (int off = 16; off > 0; off >>= 1) {
    iou += __shfl_xor(iou, off);
    bce += __shfl_xor(bce, off);
  }
  __shared__ float r2[32][2];
  const int lane = tid & 31, wave = tid >> 5;
  if (lane == 0) { r2[wave][0] = iou; r2[wave][1] = bce; }
  __syncthreads();
  if (tid < 32) {
    float si = r2[lane][0], sc = r2[lane][1];
#pragma unroll
    for (int off = 16; off > 0; off >>= 1) {
      si += __shfl_xor(si, off);
      sc += __shfl_xor(sc, off);
    }
    if (lane == 0) {
      float box_loss  = 1.f - si * inv_bn;
      float conf_loss = sc * inv_bn;
      float Sb = ws[0], Spt = ws[1], Sp = ws[2], St = ws[3];
      float bcem = Sb / ntotf;
      float dice = (2.f * Spt + 1e-5f) / (Sp + St + 1e-5f);
      float tml  = 0.5f * bcem + 0.5f * (1.f - dice);
      float total = tml + box_loss + 0.5f * conf_loss;
      out[0] = total; out[1] = tml; out[2] = box_loss; out[3] = conf_loss;
    }
  }
}

__global__ void zero_ws(float* ws) {
  if (threadIdx.x < 4) ws[threadIdx.x] = 0.f;
}

extern "C" void kernel_launch(void* const* d_in, const int* in_sizes, int n_in,
                              void* d_out, int out_size, void* d_ws, size_t ws_size,
                              hipStream_t stream) {
  const float* p     = (const float*)d_in[0];   // text_map_pred   (16,1,640,640)
  const float* t     = (const float*)d_in[1];   // text_map_target (16,1,640,640)
  const float* conf  = (const float*)d_in[2];   // confidence      (16,1,640,640)
  const float* bbox  = (const float*)d_in[3];   // bbox_coords     (16,4,640,640)
  const float* boxes = (const float*)d_in[4];   // boxes           (16,64,5)
  float* out = (float*)d_out;
  float* ws  = (float*)d_ws;

  long long ntot = in_sizes[0];                 // 6,553,600
  long long nvec = ntot / 4;                    // float4 count
  int bn = in_sizes[4] / 5;                     // 1024 boxes

  zero_ws<<<1, 32, 0, stream>>>(ws);
  tmap_kernel<<<NBLK, TPB, 0, stream>>>(p, t, ws, nvec);
  box_kernel<<<1, 1024, 0, stream>>>(conf, bbox, boxes, ws, out,
                                     (float)ntot, 1.0f / (float)bn);
}